// CustomLoss_74947179315439
// MI455X (gfx1250) — compile-verified
//
#include <hip/hip_runtime.h>
#include <hip/hip_bf16.h>

// ---------------------------------------------------------------------------
// BCE-with-logits loss reduction, MI455X (gfx1250, wave32).
//
// Pass 1 (bce_partial): 1024 blocks x 256 threads, grid-stride over float4 /
//   int4 groups with non-temporal b128 loads (268 MB streamed once; > 192 MB
//   L2, so NT avoids cache thrash). Per-thread f32 accumulation + int positive
//   count, wave32 shuffle tree, LDS cross-wave tree, one (float, int) partial
//   per block into d_ws.
//
// Pass 2 (bce_final): single 256-thread block. Reduces 1024 partials to 32
//   lane partials, then uses V_WMMA_F32_16X16X4_F32 (full fp32 precision) with
//   a ones B-matrix as the cross-lane reduction, plus one shuffle to combine
//   the two half-wave column sums. Integer count reduced exactly via shuffles.
// ---------------------------------------------------------------------------

typedef float v4f __attribute__((ext_vector_type(4)));
typedef int   v4i __attribute__((ext_vector_type(4)));
typedef float v2f __attribute__((ext_vector_type(2)));
typedef float v8f __attribute__((ext_vector_type(8)));

#define NBLOCKS  1024
#define NTHREADS 256

__device__ __forceinline__ float bce_term(float x, int lab, int& pos) {
  // y*log_sigmoid(x) + (1-y)*log_sigmoid(-x) == y*x - max(x,0) - log(1+exp(-|x|))
  const bool p = (lab != 0);
  pos += p ? 1 : 0;
  const float y = p ? 1.0f : 0.0f;
  const float e = __expf(-fabsf(x));           // v_exp_f32
  return fmaf(y, x, -fmaxf(x, 0.0f)) - __logf(1.0f + e);  // v_log_f32
}

__global__ void __launch_bounds__(NTHREADS)
bce_partial(const float* __restrict__ pred, const int* __restrict__ lab,
            float* __restrict__ ploss, int* __restrict__ ppos, int n) {
  const int tid    = threadIdx.x;
  const int gid    = blockIdx.x * NTHREADS + tid;
  const int stride = gridDim.x * NTHREADS;
  const int n4     = n >> 2;

  const v4f* __restrict__ p4 = (const v4f*)pred;
  const v4i* __restrict__ l4 = (const v4i*)lab;

  float loss = 0.0f;
  int   pos  = 0;

  #pragma unroll 4
  for (int i = gid; i < n4; i += stride) {
    const v4f x = __builtin_nontemporal_load(p4 + i);  // global_load_b128 (NT)
    const v4i y = __builtin_nontemporal_load(l4 + i);  // global_load_b128 (NT)
    loss += bce_term(x.x, y.x, pos);
    loss += bce_term(x.y, y.y, pos);
    loss += bce_term(x.z, y.z, pos);
    loss += bce_term(x.w, y.w, pos);
  }
  if (gid == 0) {  // scalar tail (n not multiple of 4)
    for (int i = n4 << 2; i < n; ++i) loss += bce_term(pred[i], lab[i], pos);
  }

  // wave32 shuffle tree
  #pragma unroll
  for (int off = 16; off > 0; off >>= 1) {
    loss += __shfl_down(loss, off, 32);
    pos  += __shfl_down(pos,  off, 32);
  }

  __shared__ float wl[NTHREADS / 32];
  __shared__ int   wp[NTHREADS / 32];
  const int wave = tid >> 5, lane = tid & 31;
  if (lane == 0) { wl[wave] = loss; wp[wave] = pos; }
  __syncthreads();
  if (tid == 0) {
    float L = 0.0f; int P = 0;
    #pragma unroll
    for (int w = 0; w < NTHREADS / 32; ++w) { L += wl[w]; P += wp[w]; }
    ploss[blockIdx.x] = L;
    ppos[blockIdx.x]  = P;
  }
}

__global__ void __launch_bounds__(NTHREADS)
bce_final(const float* __restrict__ ploss, const int* __restrict__ ppos,
          float* __restrict__ out, int nblocks, int n) {
  const int tid = threadIdx.x;
  __shared__ float sl[NTHREADS];
  __shared__ int   sp[NTHREADS];

  float L = 0.0f; int P = 0;
  for (int i = tid; i < nblocks; i += NTHREADS) { L += ploss[i]; P += ppos[i]; }
  sl[tid] = L; sp[tid] = P;
  __syncthreads();

  if (tid < 32) {  // wave-uniform: wave 0 enters with EXEC all-ones
    float p = 0.0f; int q = 0;
    #pragma unroll
    for (int i = 0; i < NTHREADS / 32; ++i) {
      p += sl[tid * (NTHREADS / 32) + i];
      q += sp[tid * (NTHREADS / 32) + i];
    }

    // f32 WMMA reduction of the 32 lane partials (exact fp32 math):
    // A(16x4): VGPR0 = {K0 for lanes 0-15, K2 for lanes 16-31} = p, VGPR1 = 0.
    // B(4x16) = ones  =>  D[m][n] = p[m] + p[m+16]  (replicated across n).
    // Column 0 lives in lane 0 (M=0..7) and lane 16 (M=8..15) across 8 VGPRs.
    v2f a; a.x = p;    a.y = 0.0f;
    v2f b; b.x = 1.0f; b.y = 1.0f;
    v8f c = {0.f, 0.f, 0.f, 0.f, 0.f, 0.f, 0.f, 0.f};
    v8f d = __builtin_amdgcn_wmma_f32_16x16x4_f32(
        /*neg_a=*/false, a, /*neg_b=*/false, b,
        /*c_mod=*/(short)0, c, /*reuse_a=*/false, /*reuse_b=*/false);
    float s = d[0] + d[1] + d[2] + d[3] + d[4] + d[5] + d[6] + d[7];
    const float tot = s + __shfl(s, 16, 32);   // lane0 halves + lane16 halves

    // exact integer count reduction
    #pragma unroll
    for (int off = 16; off > 0; off >>= 1) q += __shfl_down(q, off, 32);

    if (tid == 0) {
      const float positive = (float)q;
      const float negative = (float)(n - q);
      const float scale = -1.0f / ((1.0f + negative) * positive);
      out[0] = scale * tot;
    }
  }
}

extern "C" void kernel_launch(void* const* d_in, const int* in_sizes, int n_in,
                              void* d_out, int out_size, void* d_ws, size_t ws_size,
                              hipStream_t stream) {
  const float* pred = (const float*)d_in[0];
  const int*   lab  = (const int*)d_in[1];
  const int    n    = in_sizes[0];

  float* ploss = (float*)d_ws;
  int*   ppos  = (int*)((char*)d_ws + NBLOCKS * sizeof(float));

  bce_partial<<<NBLOCKS, NTHREADS, 0, stream>>>(pred, lab, ploss, ppos, n);
  bce_final<<<1, NTHREADS, 0, stream>>>(ploss, ppos, (float*)d_out, NBLOCKS, n);
}